// Rwkv6SelfAttention0_35390530519757
// MI455X (gfx1250) — compile-verified
//
#include <hip/hip_runtime.h>
#include <hip/hip_bf16.h>
#include <math.h>

// RWKV6 forward for gfx1250 (MI455X), fp32 end-to-end.
// Matrix ops use V_WMMA_F32_16X16X4_F32 (native fp32 WMMA, wave32).
// Each wave computes a 32x64 output tile (MT=2 x NT=4 = 8 independent
// accumulators): B fragments are shared across the two M-tiles (halves L2
// B-panel traffic), A fragments are reused across the four N-tiles, and all
// fragment loads of a k-step are issued before the 8 independent v_wmma ops.
//
// Shapes: T=1024 B=4 C=1024 H=16 N=64 E=32; tokens TB=4096.
// d_out = [ out (T,B,C) | new_shifted (B,C) | new_state (B,H,N,N) ]
// ws layout (floats): XX(4M) XXX(4M, reused as GOUT) M5(20M) MT(0.64M)
//                     R K V G W (5x4M) TD1(0.125M) WKVO(4M)  ~= 222 MB

#define Tdim 1024
#define Bdim 4
#define Cdim 1024
#define Hdim 16
#define Ndim 64
#define Edim 32
#define TBtok (Tdim * Bdim)          // 4096
#define TBC   ((size_t)TBtok * Cdim) // 4194304

typedef float v2f __attribute__((ext_vector_type(2)));
typedef float v8f __attribute__((ext_vector_type(8)));

// ---------------------------------------------------------------------------
// Epilogue modes for the generic WMMA GEMM
//   0: identity   1: tanh   2: w = exp(-exp(x + colbias[n]))   3: swish
// ---------------------------------------------------------------------------
__device__ __forceinline__ float gemm_epilogue(float x, int mode, float cb) {
    if (mode == 1) return tanhf(x);
    if (mode == 2) { float t = x + cb; return __expf(-__expf(t)); }
    if (mode == 3) return x / (1.0f + __expf(-x));
    return x;
}

// ---------------------------------------------------------------------------
// Generic fp32 WMMA GEMM: D(MxN) = epilogue(A(MxK) @ B(KxN)).
// MT = 16-row tiles per wave, NT = 16-col tiles per wave.
// A-frag layout (32-bit A 16x4): lane L(<16): v={A[L][k0],A[L][k0+1]};
//   lane L(>=16): v={A[L-16][k0+2],A[L-16][k0+3]}.
// B-frag (4x16 row striped across lanes): v0=B[k0+2g][n], v1=B[k0+2g+1][n],
//   g=lane>>4, n=lane&15.
// D layout: VGPR i holds D[tm*16 + i + 8*(lane>>4)][tn*16 + (lane&15)].
// ---------------------------------------------------------------------------
template <int MT, int NT>
__global__ void rwkv6_wmma_gemm_f32(const float* __restrict__ A,
                                    const float* __restrict__ B,
                                    float* __restrict__ D,
                                    int M, int N, int K,
                                    int lda, int ldb, int ldd,
                                    int mode, const float* __restrict__ colbias) {
    const int wavesPerBlock = blockDim.x >> 5;
    const int wid  = blockIdx.x * wavesPerBlock + (threadIdx.x >> 5);
    const int lane = threadIdx.x & 31;
    const int ngrp = N / (16 * NT);          // tile groups along N
    const int tiles = (M / (16 * MT)) * ngrp;
    if (wid >= tiles) return;                // wave-uniform: EXEC stays all-ones

    const int tm = wid / ngrp;
    const int tg = wid - tm * ngrp;
    const int row0 = tm * (16 * MT) + (lane & 15);
    const int col0 = tg * (16 * NT) + (lane & 15);
    const int kg   = (lane >> 4) << 1;       // 0 or 2

    const float* __restrict__ arow[MT];
#pragma unroll
    for (int i = 0; i < MT; ++i) {
        arow[i] = A + (size_t)(row0 + 16 * i) * lda + kg;
        __builtin_prefetch(arow[i], 0, 3);   // global_prefetch_b8, warm L2
    }
    __builtin_prefetch(B + col0, 0, 3);

    v8f acc[MT][NT];
#pragma unroll
    for (int i = 0; i < MT; ++i)
#pragma unroll
        for (int j = 0; j < NT; ++j)
            acc[i][j] = (v8f){0.f,0.f,0.f,0.f,0.f,0.f,0.f,0.f};

#pragma unroll 4
    for (int k0 = 0; k0 < K; k0 += 4) {
        // Gather all fragments for this k-step first (batch the VMEM),
        // then issue MT*NT independent WMMAs.
        v2f af[MT];
#pragma unroll
        for (int i = 0; i < MT; ++i) {
            af[i].x = arow[i][k0];
            af[i].y = arow[i][k0 + 1];
        }
        const float* __restrict__ b0 = B + (size_t)(k0 + kg) * ldb + col0;
        const float* __restrict__ b1 = b0 + ldb;
        v2f bf[NT];
#pragma unroll
        for (int j = 0; j < NT; ++j) {
            bf[j].x = b0[16 * j];
            bf[j].y = b1[16 * j];
        }
#pragma unroll
        for (int i = 0; i < MT; ++i)
#pragma unroll
            for (int j = 0; j < NT; ++j)
                acc[i][j] = __builtin_amdgcn_wmma_f32_16x16x4_f32(
                    /*neg_a=*/false, af[i], /*neg_b=*/false, bf[j],
                    /*c_mod=*/(short)0, acc[i][j],
                    /*reuse_a=*/false, /*reuse_b=*/false);
    }

#pragma unroll
    for (int i = 0; i < MT; ++i) {
        const int mbase = tm * (16 * MT) + 16 * i + ((lane >> 4) << 3);
#pragma unroll
        for (int j = 0; j < NT; ++j) {
            const int col = col0 + 16 * j;
            const float cb = (mode == 2) ? colbias[col] : 0.0f;
#pragma unroll
            for (int e = 0; e < 8; ++e)
                D[(size_t)(mbase + e) * ldd + col] =
                    gemm_epilogue(acc[i][j][e], mode, cb);
        }
    }
}

// ---------------------------------------------------------------------------
// Token shift: xx = prev - x ; xxx = x + xx * time_maa_x ; new_shifted = x[T-1]
// ---------------------------------------------------------------------------
__global__ void rwkv6_prep_kernel(const float* __restrict__ x,
                                  const float* __restrict__ shifted,
                                  const float* __restrict__ tmaax,
                                  float* __restrict__ xx,
                                  float* __restrict__ xxx,
                                  float* __restrict__ shifted_out) {
    size_t j = (size_t)blockIdx.x * blockDim.x + threadIdx.x;
    if (j >= TBC) return;
    int bc = (int)(j % (Bdim * Cdim));
    int t  = (int)(j / (Bdim * Cdim));
    int c  = (int)(j % Cdim);
    float xv = x[j];
    float pv = (t == 0) ? shifted[bc] : x[j - (size_t)Bdim * Cdim];
    float d  = pv - xv;
    xx[j]  = d;
    xxx[j] = xv + d * tmaax[c];
    if (t == Tdim - 1) shifted_out[bc] = xv;
}

// ---------------------------------------------------------------------------
// Mix (in place over m5): m5[i] <- x + xx * (time_maa_i + m5[i]),
// i order = (w, k, v, r, g) matching the reference's m[0..4].
// ---------------------------------------------------------------------------
__global__ void rwkv6_mix_kernel(const float* __restrict__ x,
                                 const float* __restrict__ xx,
                                 float* __restrict__ m5,
                                 const float* __restrict__ tw,
                                 const float* __restrict__ tk,
                                 const float* __restrict__ tv,
                                 const float* __restrict__ tr,
                                 const float* __restrict__ tg) {
    size_t j = (size_t)blockIdx.x * blockDim.x + threadIdx.x;
    if (j >= 5 * TBC) return;
    int i = (int)(j / TBC);
    size_t rem = j % TBC;
    int c = (int)(rem % Cdim);
    const float* tm = (i == 0) ? tw : (i == 1) ? tk : (i == 2) ? tv
                                                : (i == 3) ? tr : tg;
    m5[j] = x[rem] + xx[rem] * (tm[c] + m5[j]);
}

// ---------------------------------------------------------------------------
// WKV recurrence. One block per (b,h); 64 threads; thread m keeps state
// column s[n] = state[b,h,n,m] in VGPRs. r/k/w broadcast via LDS per step.
//   a[n]   = k[n]*v[m]
//   out[m] = sum_n r[n]*(tf[n]*a[n] + s[n])
//   s[n]   = a[n] + w[n]*s[n]
// ---------------------------------------------------------------------------
__global__ void rwkv6_wkv_kernel(const float* __restrict__ r,
                                 const float* __restrict__ k,
                                 const float* __restrict__ v,
                                 const float* __restrict__ w,
                                 const float* __restrict__ tf,
                                 const float* __restrict__ st_in,
                                 float* __restrict__ out,
                                 float* __restrict__ st_out) {
    const int bh = blockIdx.x;           // 0..B*H-1
    const int b  = bh / Hdim;
    const int h  = bh - b * Hdim;
    const int m  = threadIdx.x;          // 0..63

    __shared__ float sr[Ndim], sk[Ndim], sw[Ndim], stf[Ndim];
    stf[m] = tf[h * Ndim + m];

    float s[Ndim];
    const size_t sbase = (size_t)bh * Ndim * Ndim;
#pragma unroll
    for (int n = 0; n < Ndim; ++n) s[n] = st_in[sbase + (size_t)n * Ndim + m];
    __syncthreads();

    for (int t = 0; t < Tdim; ++t) {
        const size_t base = ((size_t)t * Bdim + b) * Cdim + h * Ndim;
        sr[m] = r[base + m];
        sk[m] = k[base + m];
        sw[m] = w[base + m];
        __syncthreads();
        const float vv = v[base + m];
        float acc = 0.0f;
#pragma unroll
        for (int n = 0; n < Ndim; ++n) {
            float a = sk[n] * vv;
            acc += sr[n] * (stf[n] * a + s[n]);
            s[n] = a + sw[n] * s[n];
        }
        out[base + m] = acc;
        __syncthreads();
    }
#pragma unroll
    for (int n = 0; n < Ndim; ++n) st_out[sbase + (size_t)n * Ndim + m] = s[n];
}

// ---------------------------------------------------------------------------
// GroupNorm with stats over (B, N) per (t, h) [matches reference axes (1,3)],
// eps = 1e-5 * 8^2, then scale/bias per channel and multiply by g.
// One block of 256 per (t,h): thread j -> (b = j>>6, n = j&63).
// ---------------------------------------------------------------------------
__global__ void rwkv6_gnorm_kernel(const float* __restrict__ wkv,
                                   const float* __restrict__ g,
                                   const float* __restrict__ scale,
                                   const float* __restrict__ bias,
                                   float* __restrict__ gout) {
    const int th = blockIdx.x;           // t*H + h
    const int t  = th / Hdim;
    const int h  = th - t * Hdim;
    const int j  = threadIdx.x;          // 0..255
    const int b  = j >> 6;
    const int n  = j & 63;
    const int c  = h * Ndim + n;
    const size_t idx = ((size_t)t * Bdim + b) * Cdim + c;

    const float x = wkv[idx];
    __shared__ float ssum[256], ssq[256];
    ssum[j] = x;
    ssq[j]  = x * x;
    __syncthreads();
    for (int off = 128; off > 0; off >>= 1) {
        if (j < off) { ssum[j] += ssum[j + off]; ssq[j] += ssq[j + off]; }
        __syncthreads();
    }
    const float mean = ssum[0] * (1.0f / 256.0f);
    const float var  = ssq[0] * (1.0f / 256.0f) - mean * mean;
    const float xn   = (x - mean) * rsqrtf(var + 6.4e-4f);
    gout[idx] = (xn * scale[c] + bias[c]) * g[idx];
}

// ---------------------------------------------------------------------------
static inline void launch_gemm(const float* A, const float* B, float* D,
                               int M, int N, int K, int lda, int ldb, int ldd,
                               int mode, const float* colbias, hipStream_t s) {
    if ((N & 63) == 0) {                              // 32x64 tiles per wave
        const int tiles  = (M / 32) * (N / 64);
        const int blocks = (tiles + 7) / 8;           // 8 waves / block
        rwkv6_wmma_gemm_f32<2, 4><<<dim3(blocks), dim3(256), 0, s>>>(
            A, B, D, M, N, K, lda, ldb, ldd, mode, colbias);
    } else {                                          // narrow N (160, 32)
        const int tiles  = (M / 32) * (N / 16);
        const int blocks = (tiles + 7) / 8;
        rwkv6_wmma_gemm_f32<2, 1><<<dim3(blocks), dim3(256), 0, s>>>(
            A, B, D, M, N, K, lda, ldb, ldd, mode, colbias);
    }
}

extern "C" void kernel_launch(void* const* d_in, const int* in_sizes, int n_in,
                              void* d_out, int out_size, void* d_ws, size_t ws_size,
                              hipStream_t stream) {
    const float* inputs  = (const float*)d_in[0];
    const float* shifted = (const float*)d_in[1];
    const float* state   = (const float*)d_in[2];
    const float* tmaa_x  = (const float*)d_in[3];
    const float* tmaa_w  = (const float*)d_in[4];
    const float* tmaa_k  = (const float*)d_in[5];
    const float* tmaa_v  = (const float*)d_in[6];
    const float* tmaa_r  = (const float*)d_in[7];
    const float* tmaa_g  = (const float*)d_in[8];
    const float* w1      = (const float*)d_in[9];   // (C, 5E)
    const float* w2      = (const float*)d_in[10];  // (5, E, C)
    const float* Wr      = (const float*)d_in[11];
    const float* Wk      = (const float*)d_in[12];
    const float* Wv      = (const float*)d_in[13];
    const float* Wg      = (const float*)d_in[14];
    const float* Wo      = (const float*)d_in[15];
    const float* dw1     = (const float*)d_in[16];  // (C, E)
    const float* dw2     = (const float*)d_in[17];  // (E, C)
    const float* tdecay  = (const float*)d_in[18];  // (1,1,C)
    const float* tfaaaa  = (const float*)d_in[19];  // (H, N)
    const float* gnsc    = (const float*)d_in[20];
    const float* gnbi    = (const float*)d_in[21];
    (void)in_sizes; (void)n_in; (void)out_size; (void)ws_size;

    float* out_main    = (float*)d_out;                     // (T,B,C)
    float* out_shifted = out_main + TBC;                    // (B,C)
    float* out_state   = out_shifted + (size_t)Bdim * Cdim; // (B,H,N,N)

    float* ws   = (float*)d_ws;
    float* XX   = ws;                          // TBC
    float* XXX  = XX + TBC;                    // TBC (reused as GOUT later)
    float* M5   = XXX + TBC;                   // 5*TBC  (w,k,v,r,g)
    float* MT   = M5 + 5 * TBC;                // TB * 5E
    float* RB   = MT + (size_t)TBtok * 5 * Edim;
    float* KB   = RB + TBC;
    float* VB   = KB + TBC;
    float* GB   = VB + TBC;
    float* WB   = GB + TBC;
    float* TD1  = WB + TBC;                    // TB * E
    float* WKVO = TD1 + (size_t)TBtok * Edim;  // TBC
    float* GOUT = XXX;                         // XXX dead after MT GEMM

    const int ew_blocks  = (int)((TBC + 255) / 256);
    const int mix_blocks = (int)((5 * TBC + 255) / 256);

    // 1) Token shift / xxx / new_shifted
    rwkv6_prep_kernel<<<dim3(ew_blocks), dim3(256), 0, stream>>>(
        inputs, shifted, tmaa_x, XX, XXX, out_shifted);

    // 2) MT = tanh(XXX @ w1)                       (4096 x 160, K=1024)
    launch_gemm(XXX, w1, MT, TBtok, 5 * Edim, Cdim, Cdim, 5 * Edim, 5 * Edim,
                1, nullptr, stream);

    // 3) m_i = MT[:, i*E:(i+1)*E] @ w2[i]          (4096 x 1024, K=32) x5
    for (int i = 0; i < 5; ++i)
        launch_gemm(MT + i * Edim, w2 + (size_t)i * Edim * Cdim, M5 + (size_t)i * TBC,
                    TBtok, Cdim, Edim, 5 * Edim, Cdim, Cdim, 0, nullptr, stream);

    // 4) Mixed inputs (in place over M5): xw,xk,xv,xr,xg
    rwkv6_mix_kernel<<<dim3(mix_blocks), dim3(256), 0, stream>>>(
        inputs, XX, M5, tmaa_w, tmaa_k, tmaa_v, tmaa_r, tmaa_g);

    // 5) Decay MLP: TD1 = tanh(xw @ dw1); WB = exp(-exp(TD1 @ dw2 + time_decay))
    launch_gemm(M5 + 0 * TBC, dw1, TD1, TBtok, Edim, Cdim, Cdim, Edim, Edim,
                1, nullptr, stream);
    launch_gemm(TD1, dw2, WB, TBtok, Cdim, Edim, Edim, Cdim, Cdim,
                2, tdecay, stream);

    // 6) Big projections (4096 x 1024, K=1024)
    launch_gemm(M5 + 3 * TBC, Wr, RB, TBtok, Cdim, Cdim, Cdim, Cdim, Cdim, 0, nullptr, stream);
    launch_gemm(M5 + 1 * TBC, Wk, KB, TBtok, Cdim, Cdim, Cdim, Cdim, Cdim, 0, nullptr, stream);
    launch_gemm(M5 + 2 * TBC, Wv, VB, TBtok, Cdim, Cdim, Cdim, Cdim, Cdim, 0, nullptr, stream);
    launch_gemm(M5 + 4 * TBC, Wg, GB, TBtok, Cdim, Cdim, Cdim, Cdim, Cdim, 3, nullptr, stream); // swish

    // 7) WKV scan (serial over T) + final state
    rwkv6_wkv_kernel<<<dim3(Bdim * Hdim), dim3(Ndim), 0, stream>>>(
        RB, KB, VB, WB, tfaaaa, state, WKVO, out_state);

    // 8) GroupNorm over (B,N) per (t,h), scale/bias, multiply by g
    rwkv6_gnorm_kernel<<<dim3(Tdim * Hdim), dim3(256), 0, stream>>>(
        WKVO, GB, gnsc, gnbi, GOUT);

    // 9) out = GOUT @ Wo                           (4096 x 1024, K=1024)
    launch_gemm(GOUT, Wo, out_main, TBtok, Cdim, Cdim, Cdim, Cdim, Cdim,
                0, nullptr, stream);
}